// DGCNN_26027501814374
// MI455X (gfx1250) — compile-verified
//
#include <hip/hip_runtime.h>
#include <math.h>

// ---------------------------------------------------------------------------
// DGCNN forward for MI455X (gfx1250, wave32). All dense GEMMs run on the fp32
// tensor path V_WMMA_F32_16X16X4_F32 (exact precision vs the fp32 reference).
// ---------------------------------------------------------------------------

typedef float v2f __attribute__((ext_vector_type(2)));
typedef float v8f __attribute__((ext_vector_type(8)));

#define TILE 32
#define KNN_K 20

// ------------------------- generic WMMA fp32 GEMM ---------------------------
// C[M,Nn] = act(A[M,Kd] @ W[Kd,Nn] + bias). Strided A/C views support the
// column-slice / concat-write patterns of the reference without copies.
__global__ __launch_bounds__(128)
void gemm_wmma_f32(const float* __restrict__ A, int lda,
                   const float* __restrict__ W, int ldw,
                   const float* __restrict__ bias,
                   float* __restrict__ C, int ldc,
                   int M, int Kd, int Nn, int act)
{
    __shared__ float As[TILE][TILE + 1];
    __shared__ float Bs[TILE][TILE + 1];

    const int tid  = threadIdx.x;
    const int lane = tid & 31;
    const int wave = tid >> 5;          // 4 waves: 2x2 arrangement of 16x16 tiles
    const int wm = wave >> 1, wn = wave & 1;
    const int rowBase = blockIdx.y * TILE;
    const int colBase = blockIdx.x * TILE;

    v8f acc = {};

    for (int k0 = 0; k0 < Kd; k0 += TILE) {
        // cooperative stage of A (32xK) and B (Kx32) tiles, zero-padded
        for (int t = tid; t < TILE * TILE; t += 128) {
            int r = t >> 5, c = t & 31;
            int gr = rowBase + r, gk = k0 + c;
            As[r][c] = (gr < M && gk < Kd) ? A[(size_t)gr * lda + gk] : 0.0f;
            int gk2 = k0 + r, gc = colBase + c;
            Bs[r][c] = (gk2 < Kd && gc < Nn) ? W[(size_t)gk2 * ldw + gc] : 0.0f;
        }
        // CDNA5 prefetch of the next K-tile (global_prefetch_b8)
        if (tid < 32) {
            int nr = rowBase + tid, nk = k0 + TILE;
            if (nr < M && nk < Kd) __builtin_prefetch(&A[(size_t)nr * lda + nk], 0, 1);
            int nk2 = k0 + TILE + tid;
            if (nk2 < Kd && colBase < Nn) __builtin_prefetch(&W[(size_t)nk2 * ldw + colBase], 0, 1);
        }
        __syncthreads();

        const int kb = (lane >> 4) << 1;       // lanes 0-15: K 0,1 ; 16-31: K 2,3
        const int mr = wm * 16 + (lane & 15);
        const int nc = wn * 16 + (lane & 15);
#pragma unroll
        for (int kk = 0; kk < TILE; kk += 4) {
            v2f a, b;
            a.x = As[mr][kk + kb];     a.y = As[mr][kk + kb + 1];
            b.x = Bs[kk + kb][nc];     b.y = Bs[kk + kb + 1][nc];
            acc = __builtin_amdgcn_wmma_f32_16x16x4_f32(false, a, false, b,
                                                        (short)0, acc, false, false);
        }
        __syncthreads();
    }

    // epilogue: VGPR r holds rows (r, r+8) split across lane halves
    const int gc = colBase + wn * 16 + (lane & 15);
    const int rhalf = (lane >> 4) * 8;
#pragma unroll
    for (int r = 0; r < 8; ++r) {
        int gr = rowBase + wm * 16 + r + rhalf;
        if (gr < M && gc < Nn) {
            float v = acc[r] + (bias ? bias[gc] : 0.0f);
            if (act == 1) v = v > 0.0f ? v : 0.0f;
            C[(size_t)gr * ldc + gc] = v;
        }
    }
}

// ------------------------------ KNN (top-20) --------------------------------
// One block per query row: distances into LDS, then 20 argmin sweeps.
__global__ __launch_bounds__(256)
void knn_kernel(const float* __restrict__ X, const int* __restrict__ batch,
                int N, int D, int* __restrict__ idxOut)
{
    __shared__ float dist[8192];
    __shared__ float xi[128];
    __shared__ float rv[256];
    __shared__ int   ri[256];

    const int i = blockIdx.x;
    const int tid = threadIdx.x;

    for (int d = tid; d < D; d += 256) xi[d] = X[(size_t)i * D + d];
    __syncthreads();

    const int bi = batch[i];
    for (int j = tid; j < N; j += 256) {
        const float* xj = X + (size_t)j * D;
        float accd = 0.0f;
        for (int d = 0; d < D; ++d) {
            float df = xj[d] - xi[d];
            accd = fmaf(df, df, accd);
        }
        dist[j] = (batch[j] == bi) ? accd : INFINITY;
    }
    __syncthreads();

    for (int k = 0; k < KNN_K; ++k) {
        float best = INFINITY; int bj = 0;
        for (int j = tid; j < N; j += 256) {
            float v = dist[j];
            if (v < best) { best = v; bj = j; }
        }
        rv[tid] = best; ri[tid] = bj;
        __syncthreads();
        for (int s = 128; s > 0; s >>= 1) {
            if (tid < s && rv[tid + s] < rv[tid]) { rv[tid] = rv[tid + s]; ri[tid] = ri[tid + s]; }
            __syncthreads();
        }
        if (tid == 0) { idxOut[(size_t)i * KNN_K + k] = ri[0]; dist[ri[0]] = INFINITY; }
        __syncthreads();
    }
}

// ------------------------------- EdgeConv -----------------------------------
// out[i] = max_e ( relu([x_i, x_nbr-x_i] @ W1 + b1) @ W2 + b2 ).
// One block per node, one thread per output channel, K=20 accumulators in regs.
__global__ __launch_bounds__(256)
void edgeconv_kernel(const float* __restrict__ X, int Din,
                     const int* __restrict__ idx,
                     const float* __restrict__ W1, const float* __restrict__ b1,
                     const float* __restrict__ W2, const float* __restrict__ b2,
                     int H, float* __restrict__ Out, int ldc)
{
    extern __shared__ float smem[];
    float* ef = smem;                       // KNN_K * 2*Din
    float* h  = smem + (size_t)KNN_K * 2 * Din; // KNN_K * H

    const int i = blockIdx.x;
    const int tid = threadIdx.x;
    const int D2 = 2 * Din;

    for (int t = tid; t < KNN_K * Din; t += H) {
        int e = t / Din, d = t - e * Din;
        float xc = X[(size_t)i * Din + d];
        float nb = X[(size_t)idx[(size_t)i * KNN_K + e] * Din + d];
        ef[(size_t)e * D2 + d] = xc;
        ef[(size_t)e * D2 + Din + d] = nb - xc;
    }
    __syncthreads();

    const int o = tid;
    float a[KNN_K];
#pragma unroll
    for (int e = 0; e < KNN_K; ++e) a[e] = b1[o];
    for (int k = 0; k < D2; ++k) {
        float w = W1[(size_t)k * H + o];
#pragma unroll
        for (int e = 0; e < KNN_K; ++e) a[e] = fmaf(ef[(size_t)e * D2 + k], w, a[e]);
    }
#pragma unroll
    for (int e = 0; e < KNN_K; ++e) h[(size_t)e * H + o] = a[e] > 0.0f ? a[e] : 0.0f;
    __syncthreads();

#pragma unroll
    for (int e = 0; e < KNN_K; ++e) a[e] = b2[o];
    for (int k = 0; k < H; ++k) {
        float w = W2[(size_t)k * H + o];
#pragma unroll
        for (int e = 0; e < KNN_K; ++e) a[e] = fmaf(h[(size_t)e * H + k], w, a[e]);
    }
    float mx = a[0];
#pragma unroll
    for (int e = 1; e < KNN_K; ++e) mx = fmaxf(mx, a[e]);
    Out[(size_t)i * ldc + o] = mx;
}

// --------------------------------- GAT --------------------------------------
__global__ void build_edges(const int* __restrict__ ei, int E, int N,
                            int* __restrict__ src, int* __restrict__ dst)
{
    int e = blockIdx.x * blockDim.x + threadIdx.x;
    int T = E + N;
    if (e >= T) return;
    if (e < E) { src[e] = ei[e]; dst[e] = ei[E + e]; }
    else       { src[e] = e - E; dst[e] = e - E; }
}

__global__ void gat_lsld(const float* __restrict__ Z,
                         const float* __restrict__ as_, const float* __restrict__ ad_,
                         int N, int Cc, float* __restrict__ ls, float* __restrict__ ld)
{
    int t = blockIdx.x * blockDim.x + threadIdx.x;   // N*8 items
    if (t >= N * 8) return;
    int n = t >> 3, hh = t & 7;
    const float* z = Z + (size_t)n * 8 * Cc + (size_t)hh * Cc;
    float s0 = 0.0f, s1 = 0.0f;
    for (int c = 0; c < Cc; ++c) {
        s0 = fmaf(z[c], as_[hh * Cc + c], s0);
        s1 = fmaf(z[c], ad_[hh * Cc + c], s1);
    }
    ls[t] = s0; ld[t] = s1;
}

__global__ void fill_f32(float* __restrict__ p, float v, size_t n)
{
    size_t i = (size_t)blockIdx.x * blockDim.x + threadIdx.x;
    if (i < n) p[i] = v;
}

__device__ inline void atomicMaxF(float* addr, float val)
{
    // sign-split trick: non-negative -> signed int max; negative -> unsigned min
    if (__float_as_int(val) >= 0) atomicMax((int*)addr, __float_as_int(val));
    else                          atomicMin((unsigned int*)addr, __float_as_uint(val));
}

__global__ void gat_logit_max(const int* __restrict__ src, const int* __restrict__ dst,
                              const float* __restrict__ ls, const float* __restrict__ ld,
                              int T, float* __restrict__ logit, float* __restrict__ m)
{
    int e = blockIdx.x * blockDim.x + threadIdx.x;
    if (e >= T) return;
    int s = src[e], d = dst[e];
#pragma unroll
    for (int hh = 0; hh < 8; ++hh) {
        float l = ls[s * 8 + hh] + ld[d * 8 + hh];
        l = l > 0.0f ? l : 0.2f * l;                 // leaky_relu(0.2)
        logit[(size_t)e * 8 + hh] = l;
        atomicMaxF(&m[d * 8 + hh], l);
    }
}

__global__ void gat_exden(const int* __restrict__ dst, float* __restrict__ logit,
                          const float* __restrict__ m, float* __restrict__ den, int T)
{
    int e = blockIdx.x * blockDim.x + threadIdx.x;
    if (e >= T) return;
    int d = dst[e];
#pragma unroll
    for (int hh = 0; hh < 8; ++hh) {
        float ex = expf(logit[(size_t)e * 8 + hh] - m[d * 8 + hh]);
        logit[(size_t)e * 8 + hh] = ex;              // in-place: logit -> ex
        atomicAdd(&den[d * 8 + hh], ex);
    }
}

__global__ __launch_bounds__(256)
void gat_scatter(const int* __restrict__ src, const int* __restrict__ dst,
                 const float* __restrict__ Z, const float* __restrict__ ex,
                 const float* __restrict__ den, int HC, int Cc,
                 float* __restrict__ out)
{
    __shared__ float alpha[8];
    int e = blockIdx.x;
    int s = src[e], d = dst[e];
    if (threadIdx.x < 8) {
        int hh = threadIdx.x;
        alpha[hh] = ex[(size_t)e * 8 + hh] / (den[d * 8 + hh] + 1e-16f);
    }
    __syncthreads();
    for (int t = threadIdx.x; t < HC; t += blockDim.x) {
        int hh = t / Cc;
        atomicAdd(&out[(size_t)d * HC + t], Z[(size_t)s * HC + t] * alpha[hh]);
    }
}

__global__ void gat_finalize(const float* __restrict__ acc, const float* __restrict__ b,
                             int N, int HC, float* __restrict__ out, int ldo)
{
    size_t t = (size_t)blockIdx.x * blockDim.x + threadIdx.x;
    if (t >= (size_t)N * HC) return;
    int n = (int)(t / HC), c = (int)(t - (size_t)n * HC);
    float v = acc[t] + b[c];
    v = v > 0.0f ? v : expm1f(v);                    // elu
    out[(size_t)n * ldo + c] = v;
}

// ------------------------------ misc kernels --------------------------------
__global__ void copy2d(const float* __restrict__ src, int lds_,
                       float* __restrict__ dst, int ldd, int M, int Nc)
{
    size_t t = (size_t)blockIdx.x * blockDim.x + threadIdx.x;
    if (t >= (size_t)M * Nc) return;
    int r = (int)(t / Nc), c = (int)(t - (size_t)r * Nc);
    dst[(size_t)r * ldd + c] = src[(size_t)r * lds_ + c];
}

__global__ __launch_bounds__(256)
void final_head(const float* __restrict__ H2, const float* __restrict__ w,
                const float* __restrict__ b, int N, int Kd, float* __restrict__ out)
{
    int row = blockIdx.x * 8 + (threadIdx.x >> 5);   // one wave32 per row
    int lane = threadIdx.x & 31;
    if (row >= N) return;
    float s = 0.0f;
    for (int k = lane; k < Kd; k += 32) s = fmaf(H2[(size_t)row * Kd + k], w[k], s);
    for (int off = 16; off; off >>= 1) s += __shfl_xor(s, off, 32);
    if (lane == 0) out[row] = 1.0f / (1.0f + expf(-5.0f * (s + b[0])));
}

// ------------------------------ orchestration -------------------------------
extern "C" void kernel_launch(void* const* d_in, const int* in_sizes, int n_in,
                              void* d_out, int out_size, void* d_ws, size_t ws_size,
                              hipStream_t stream)
{
    const float* x     = (const float*)d_in[0];
    const int*   batch = (const int*)d_in[1];
    const int*   ei    = (const int*)d_in[2];

    const float *pg1W=(const float*)d_in[3],  *pg1b=(const float*)d_in[4];
    const float *pg2W=(const float*)d_in[5],  *pg2b=(const float*)d_in[6];
    const float *pg3W=(const float*)d_in[7],  *pg3b=(const float*)d_in[8];
    const float *da1W=(const float*)d_in[9],  *da1b=(const float*)d_in[10];
    const float *da2W=(const float*)d_in[11], *da2b=(const float*)d_in[12];
    const float *da3W=(const float*)d_in[13], *da3b=(const float*)d_in[14];
    const float *c1aW=(const float*)d_in[15], *c1ab=(const float*)d_in[16];
    const float *c1bW=(const float*)d_in[17], *c1bb=(const float*)d_in[18];
    const float *c2aW=(const float*)d_in[19], *c2ab=(const float*)d_in[20];
    const float *c2bW=(const float*)d_in[21], *c2bb=(const float*)d_in[22];
    const float *c3aW=(const float*)d_in[23], *c3ab=(const float*)d_in[24];
    const float *c3bW=(const float*)d_in[25], *c3bb=(const float*)d_in[26];
    const float *m1W =(const float*)d_in[27], *m1b =(const float*)d_in[28];
    const float *m2W =(const float*)d_in[29], *m2b =(const float*)d_in[30];
    const float *m3W =(const float*)d_in[31], *m3b =(const float*)d_in[32];
    const float *g1W =(const float*)d_in[33], *g1as=(const float*)d_in[34];
    const float *g1ad=(const float*)d_in[35], *g1b =(const float*)d_in[36];
    const float *g2W =(const float*)d_in[37], *g2as=(const float*)d_in[38];
    const float *g2ad=(const float*)d_in[39], *g2b =(const float*)d_in[40];
    const float *g3W =(const float*)d_in[41], *g3as=(const float*)d_in[42];
    const float *g3ad=(const float*)d_in[43], *g3b =(const float*)d_in[44];

    const int N = in_sizes[1];          // 8192
    const int E = in_sizes[2] / 2;      // 262144
    const int T = E + N;                // edges + self loops

    // ------- workspace layout (floats, then ints) -------
    float* ws = (float*)d_ws;
    size_t oXC  = 0;                                  // N*87   xc
    size_t oX3  = oXC  + (size_t)N * 87;              // N*343  x3cat
    size_t oX6  = oX3  + (size_t)N * 343;             // N*599  x6cat
    size_t oA   = oX6  + (size_t)N * 599;             // N*1024 bufA
    size_t oB   = oA   + (size_t)N * 1024;            // N*1024 bufB
    size_t oC   = oB   + (size_t)N * 1024;            // N*512  bufC
    size_t oLG  = oC   + (size_t)N * 512;             // T*8    logits/ex
    size_t oLS  = oLG  + (size_t)T * 8;
    size_t oLD  = oLS  + (size_t)N * 8;
    size_t oM   = oLD  + (size_t)N * 8;
    size_t oDEN = oM   + (size_t)N * 8;
    size_t oEnd = oDEN + (size_t)N * 8;
    float *xc = ws + oXC, *x3 = ws + oX3, *x6 = ws + oX6;
    float *bufA = ws + oA, *bufB = ws + oB, *bufC = ws + oC;
    float *logit = ws + oLG, *ls = ws + oLS, *ld = ws + oLD;
    float *mmax = ws + oM, *den = ws + oDEN;
    int* wi = (int*)(ws + oEnd);
    int* knnIdx = wi;                  // N*20
    int* srcE   = wi + (size_t)N * KNN_K;
    int* dstE   = srcE + T;

    auto blocks = [](size_t n, int b) { return dim3((unsigned)((n + b - 1) / b)); };
    auto GEMM = [&](const float* A, int lda, const float* W, int ldw, const float* b,
                    float* Cp, int ldc, int M, int Kd, int Nn, int act) {
        dim3 grid((Nn + TILE - 1) / TILE, (M + TILE - 1) / TILE);
        gemm_wmma_f32<<<grid, 128, 0, stream>>>(A, lda, W, ldw, b, Cp, ldc, M, Kd, Nn, act);
    };

    // ------- input MLPs: xc = [x[:, :39], pg(x[:,39:1063]), da(x[:,1063:2087])]
    copy2d<<<blocks((size_t)N * 39, 256), 256, 0, stream>>>(x, 2087, xc, 87, N, 39);
    GEMM(x + 39,   2087, pg1W, 512, pg1b, bufA,    512, N, 1024, 512, 1);
    GEMM(bufA,      512, pg2W,  64, pg2b, bufB,     64, N,  512,  64, 1);
    GEMM(bufB,       64, pg3W,  16, pg3b, xc + 39,  87, N,   64,  16, 0);
    GEMM(x + 1063, 2087, da1W, 256, da1b, bufA,    256, N, 1024, 256, 1);
    GEMM(bufA,      256, da2W,  64, da2b, bufB,     64, N,  256,  64, 1);
    GEMM(bufB,       64, da3W,  32, da3b, xc + 55,  87, N,   64,  32, 0);

    // ------- EdgeConv stack -------
    knn_kernel<<<N, 256, 0, stream>>>(xc, batch, N, 87, knnIdx);
    edgeconv_kernel<<<N, 64, (size_t)(KNN_K * (2 * 87) + KNN_K * 64) * 4, stream>>>(
        xc, 87, knnIdx, c1aW, c1ab, c1bW, c1bb, 64, bufA, 64);           // x1

    knn_kernel<<<N, 256, 0, stream>>>(bufA, batch, N, 64, knnIdx);
    edgeconv_kernel<<<N, 128, (size_t)(KNN_K * (2 * 64) + KNN_K * 128) * 4, stream>>>(
        bufA, 64, knnIdx, c2aW, c2ab, c2bW, c2bb, 128, bufB, 128);       // x2

    knn_kernel<<<N, 256, 0, stream>>>(bufB, batch, N, 128, knnIdx);
    edgeconv_kernel<<<N, 256, (size_t)(KNN_K * (2 * 128) + KNN_K * 256) * 4, stream>>>(
        bufB, 128, knnIdx, c3aW, c3ab, c3bW, c3bb, 256, x3, 343);        // x3[:, :256]
    copy2d<<<blocks((size_t)N * 87, 256), 256, 0, stream>>>(xc, 87, x3 + 256, 343, N, 87);

    // ------- GAT layers -------
    build_edges<<<blocks((size_t)T, 256), 256, 0, stream>>>(ei, E, N, srcE, dstE);

    auto GAT = [&](const float* Xin, int ldx, int Fin, const float* Wg,
                   const float* as_, const float* ad_, const float* bg,
                   int Cc, float* Zbuf, float* accum, float* outp, int ldo) {
        int HC = 8 * Cc;
        GEMM(Xin, ldx, Wg, HC, nullptr, Zbuf, HC, N, Fin, HC, 0);
        gat_lsld<<<blocks((size_t)N * 8, 256), 256, 0, stream>>>(Zbuf, as_, ad_, N, Cc, ls, ld);
        fill_f32<<<blocks((size_t)N * 8, 256), 256, 0, stream>>>(mmax, -INFINITY, (size_t)N * 8);
        fill_f32<<<blocks((size_t)N * 8, 256), 256, 0, stream>>>(den, 0.0f, (size_t)N * 8);
        fill_f32<<<blocks((size_t)N * HC, 256), 256, 0, stream>>>(accum, 0.0f, (size_t)N * HC);
        gat_logit_max<<<blocks((size_t)T, 256), 256, 0, stream>>>(srcE, dstE, ls, ld, T, logit, mmax);
        gat_exden<<<blocks((size_t)T, 256), 256, 0, stream>>>(dstE, logit, mmax, den, T);
        gat_scatter<<<T, 256, 0, stream>>>(srcE, dstE, Zbuf, logit, den, HC, Cc, accum);
        gat_finalize<<<blocks((size_t)N * HC, 256), 256, 0, stream>>>(accum, bg, N, HC, outp, ldo);
    };

    GAT(x3,   343,  343, g1W, g1as, g1ad, g1b, 128, bufA, bufB, bufB, 1024);   // x4 in bufB
    GAT(bufB, 1024, 1024, g2W, g2as, g2ad, g2b,  64, bufA, bufC, bufC,  512);  // x5 in bufC
    GAT(bufC,  512,  512, g3W, g3as, g3ad, g3b,  32, bufA, bufB, x6,    599);  // x6cat[:, :256]
    copy2d<<<blocks((size_t)N * 343, 256), 256, 0, stream>>>(x3, 343, x6 + 256, 599, N, 343);

    // ------- output MLP + sigmoid(5x) -------
    GEMM(x6,   599, m1W, 1024, m1b, bufA, 1024, N,  599, 1024, 1);
    GEMM(bufA, 1024, m2W, 256, m2b, bufB,  256, N, 1024,  256, 1);
    final_head<<<blocks((size_t)N, 8), 256, 0, stream>>>(bufB, m3W, m3b, N, 256, (float*)d_out);

    (void)n_in; (void)out_size; (void)ws_size;
}